// GCN_90933047591260
// MI455X (gfx1250) — compile-verified
//
#include <hip/hip_runtime.h>

// ---------------------------------------------------------------------------
// 2-layer GCN for MI455X (gfx1250, wave32).
// GEMMs via V_WMMA_F32_16X16X4_F32 (exact fp32), B tile staged in LDS,
// compile-time K/N so the whole K-loop unrolls into wmma + immediate-offset
// loads. Aggregation via f32 atomics (memory-bound, dominates runtime).
// ---------------------------------------------------------------------------

typedef __attribute__((ext_vector_type(2))) float v2f;
typedef __attribute__((ext_vector_type(8))) float v8f;

__global__ void k_zero(float* __restrict__ p, int n) {
  int i = blockIdx.x * blockDim.x + threadIdx.x;
  if (i < n) p[i] = 0.0f;
}

// in-degree count over real edges (self-loops handled analytically later)
__global__ void k_count(const int* __restrict__ ei, float* __restrict__ deg, int nE) {
  int e = blockIdx.x * blockDim.x + threadIdx.x;
  if (e < nE) atomicAdd(&deg[ei[nE + e]], 1.0f);
}

// dinv[i] = rsqrt(deg[i] + 1)   (+1 = self loop; always > 0)
__global__ void k_dinv(float* __restrict__ deg, int n) {
  int i = blockIdx.x * blockDim.x + threadIdx.x;
  if (i < n) deg[i] = rsqrtf(deg[i] + 1.0f);
}

// C[M,NCOL] = A[M,K] @ B[K,NCOL].  256 threads = 8 waves; all waves share one
// 16-column B tile staged in LDS; each wave computes one 16x16 output tile.
// Requires M % 16 == 0 (50000 = 3125*16).
// A-matrix f32 16x4 lane layout: lane l -> row (l&15), K pair (l>>4)*2.
// B mirrored: lane l -> col (l&15), K pair (l>>4)*2.
// C/D: VGPR v -> row v + 8*(l>>4), col (l&15).
template <int K, int NCOL>
__global__ void __launch_bounds__(256)
k_gemm_f32_wmma(const float* __restrict__ A, const float* __restrict__ B,
                float* __restrict__ C, int M) {
  __shared__ float BT[K * 16];
  const int tid  = threadIdx.x;
  const int lane = tid & 31;
  const int wave = tid >> 5;
  const int col0 = blockIdx.y * 16;

  // stage B tile [K x 16] into LDS, coalesced 64B rows
#pragma unroll
  for (int i = tid; i < K * 16; i += 256)
    BT[i] = B[(i >> 4) * NCOL + col0 + (i & 15)];
  __syncthreads();

  const int row0 = (blockIdx.x * 8 + wave) * 16;
  if (row0 < M) {                              // whole wave exits together
    const int half = lane >> 4;                // 0: K={0,1}, 1: K={2,3}
    const int lm   = lane & 15;
    const float* __restrict__ arow = A + (row0 + lm) * K + half * 2;
    const float* __restrict__ bp   = BT + half * 32 + lm;

    v8f c = {};
#pragma unroll
    for (int k0 = 0; k0 < K; k0 += 4) {
      v2f a = *(const v2f*)(arow + k0);        // b64 load, immediate offset
      v2f b;
      b.x = bp[k0 * 16];                       // ds_load, immediate offset
      b.y = bp[k0 * 16 + 16];
      c = __builtin_amdgcn_wmma_f32_16x16x4_f32(false, a, false, b,
                                                (short)0, c, false, false);
    }

    float* __restrict__ cc = C + (row0 + 8 * half) * NCOL + col0 + lm;
#pragma unroll
    for (int v = 0; v < 8; ++v) cc[v * NCOL] = c[v];
  }
}

// out[i,f] = z[i,f] * dinv[i]^2   (self-loop term; fully initializes buffer)
template <int F>
__global__ void k_self_init(const float* __restrict__ z, const float* __restrict__ dinv,
                            float* __restrict__ out, int n) {
  long long i = (long long)blockIdx.x * blockDim.x + threadIdx.x;
  if (i >= (long long)n * F) return;
  float di = dinv[(int)(i / F)];
  out[i] = z[i] * di * di;
}

// one wave per edge: out[dst,:] += z[src,:] * dinv[src]*dinv[dst]
template <int F>
__global__ void k_edge_agg(const float* __restrict__ z, const int* __restrict__ ei,
                           const float* __restrict__ dinv, float* __restrict__ out,
                           int nE) {
  const int lane = threadIdx.x & 31;
  const long long wid = ((long long)blockIdx.x * blockDim.x + threadIdx.x) >> 5;
  if (wid >= nE) return;
  const int s = ei[wid];
  const int d = ei[(long long)nE + wid];
  const float w = dinv[s] * dinv[d];
  constexpr int VPL = F / 32;                  // floats per lane
  const float* __restrict__ zs = z + (long long)s * F + lane * VPL;
  float* __restrict__ od = out + (long long)d * F + lane * VPL;
#pragma unroll
  for (int i = 0; i < VPL; ++i) atomicAdd(od + i, zs[i] * w);  // non-returning f32 atomic
}

template <int F>
__global__ void k_bias_relu(float* __restrict__ h, const float* __restrict__ bias, int n) {
  long long i = (long long)blockIdx.x * blockDim.x + threadIdx.x;
  if (i >= (long long)n * F) return;
  float v = h[i] + bias[(int)(i % F)];
  h[i] = v > 0.0f ? v : 0.0f;
}

static inline int cdiv(long long a, long long b) { return (int)((a + b - 1) / b); }

extern "C" void kernel_launch(void* const* d_in, const int* in_sizes, int n_in,
                              void* d_out, int out_size, void* d_ws, size_t ws_size,
                              hipStream_t stream) {
  const float* x  = (const float*)d_in[0];
  const int*   ei = (const int*)  d_in[1];   // [2, E] int32
  const float* W1 = (const float*)d_in[2];
  const float* b1 = (const float*)d_in[3];
  const float* W2 = (const float*)d_in[4];
  const float* b2 = (const float*)d_in[5];
  float* out = (float*)d_out;

  const int DIN = 256, DH1 = 128, DH2 = 64;
  const int nN = in_sizes[0] / DIN;            // 50000
  const int nE = in_sizes[1] / 2;              // 1,600,000

  // workspace layout (fp32): deg/dinv | z1 | h1 | z2   (~64.3 MB)
  float* deg = (float*)d_ws;
  float* z1  = deg + (((size_t)nN + 255) & ~(size_t)255);
  float* h1  = z1 + (size_t)nN * DH1;
  float* z2  = h1 + (size_t)nN * DH1;

  // --- normalization ---
  k_zero <<<cdiv(nN, 256), 256, 0, stream>>>(deg, nN);
  k_count<<<cdiv(nE, 256), 256, 0, stream>>>(ei, deg, nE);
  k_dinv <<<cdiv(nN, 256), 256, 0, stream>>>(deg, nN);   // deg now holds dinv
  const float* dinv = deg;

  // --- layer 1: z1 = x @ W1 ; h1 = relu(scatter(z1) + b1) ---
  k_gemm_f32_wmma<256, 128><<<dim3(cdiv(nN, 128), 128 / 16), 256, 0, stream>>>(x, W1, z1, nN);
  k_self_init<DH1><<<cdiv((long long)nN * DH1, 256), 256, 0, stream>>>(z1, dinv, h1, nN);
  k_edge_agg <DH1><<<cdiv((long long)nE * 32, 256), 256, 0, stream>>>(z1, ei, dinv, h1, nE);
  k_bias_relu<DH1><<<cdiv((long long)nN * DH1, 256), 256, 0, stream>>>(h1, b1, nN);

  // --- layer 2: z2 = h1 @ W2 ; out = relu(scatter(z2) + b2) ---
  k_gemm_f32_wmma<128, 64><<<dim3(cdiv(nN, 128), 64 / 16), 256, 0, stream>>>(h1, W2, z2, nN);
  k_self_init<DH2><<<cdiv((long long)nN * DH2, 256), 256, 0, stream>>>(z2, dinv, out, nN);
  k_edge_agg <DH2><<<cdiv((long long)nE * 32, 256), 256, 0, stream>>>(z2, ei, dinv, out, nE);
  k_bias_relu<DH2><<<cdiv((long long)nN * DH2, 256), 256, 0, stream>>>(out, b2, nN);
}